// Network24_26551487824347
// MI455X (gfx1250) — compile-verified
//
#include <hip/hip_runtime.h>
#include <cstdint>

// ---------------------------------------------------------------------------
// Elementwise "Network24": per row r
//   h_j  = sigmoid( sum_k fc1_tw[j,k,0] * x_k^fc1_power[j,k] + fc1_bias[j] )
//   out  = sigmoid( a0*h0^q0 + a1*h1^q1 + (a2*h0^q2)*(a3*h1^q3) + c )
// Memory bound: 12 B/row * 8.39M rows ~= 100 MB -> ~4.3us at 23.3 TB/s.
// Data path: CDNA5 async global->LDS double-buffer (ASYNCcnt), wave32.
// ---------------------------------------------------------------------------

struct NetP {
    float w00, w01, w10, w11;   // fc1_tw[j,k,0]
    float p00, p01, p10, p11;   // fc1_power[j,k]
    float b0, b1;               // fc1_bias
    float a0, a1, a2, a3;       // m4_tw[i,0]
    float q0, q1, q2, q3;       // m4_power
    float c;                    // m4_bias3
    int   fast;                 // all powers == 1.0f
};

__device__ __forceinline__ NetP load_params(const float* __restrict__ fc1_tw,
                                            const float* __restrict__ fc1_pw,
                                            const float* __restrict__ fc1_b,
                                            const float* __restrict__ m4_tw,
                                            const float* __restrict__ m4_pw,
                                            const float* __restrict__ m4_b) {
    NetP s;
    // fc1_tw shape (2,2,3): [j][k][0] -> (j*2+k)*3
    s.w00 = fc1_tw[0]; s.w01 = fc1_tw[3]; s.w10 = fc1_tw[6]; s.w11 = fc1_tw[9];
    s.p00 = fc1_pw[0]; s.p01 = fc1_pw[1]; s.p10 = fc1_pw[2]; s.p11 = fc1_pw[3];
    s.b0  = fc1_b[0];  s.b1  = fc1_b[1];
    // m4_tw shape (4,3): [i][0] -> i*3
    s.a0 = m4_tw[0]; s.a1 = m4_tw[3]; s.a2 = m4_tw[6]; s.a3 = m4_tw[9];
    s.q0 = m4_pw[0]; s.q1 = m4_pw[1]; s.q2 = m4_pw[2]; s.q3 = m4_pw[3];
    s.c  = m4_b[0];
    s.fast = (s.p00 == 1.0f) & (s.p01 == 1.0f) & (s.p10 == 1.0f) & (s.p11 == 1.0f) &
             (s.q0  == 1.0f) & (s.q1  == 1.0f) & (s.q2  == 1.0f) & (s.q3  == 1.0f);
    return s;
}

__device__ __forceinline__ float sigf(float z) {
    // 1/(1+e^-z): v_exp_f32 + v_rcp_f32 (<= ~2 ulp total)
    return __builtin_amdgcn_rcpf(1.0f + __expf(-z));
}

__device__ __forceinline__ float net_row(float x0, float x1, const NetP& s) {
    if (s.fast) {
        // all exponents are 1.0 -> skip all pow (uniform branch across wave)
        float h0 = sigf(fmaf(s.w00, x0, fmaf(s.w01, x1, s.b0)));
        float h1 = sigf(fmaf(s.w10, x0, fmaf(s.w11, x1, s.b1)));
        float z  = fmaf(s.a0, h0, fmaf(s.a1, h1, fmaf(s.a2 * h0, s.a3 * h1, s.c)));
        return sigf(z);
    } else {
        float h0 = sigf(fmaf(s.w00, __powf(x0, s.p00),
                        fmaf(s.w01, __powf(x1, s.p01), s.b0)));
        float h1 = sigf(fmaf(s.w10, __powf(x0, s.p10),
                        fmaf(s.w11, __powf(x1, s.p11), s.b1)));
        float s1 = s.a0 * __powf(h0, s.q0);
        float s2 = s.a1 * __powf(h1, s.q1);
        float pp = (s.a2 * __powf(h0, s.q2)) * (s.a3 * __powf(h1, s.q3));
        return sigf(s1 + s2 + pp + s.c);
    }
}

// ---------------------------------------------------------------------------
// Async double-buffered kernel: each lane stages 16B (2 rows of x) per
// iteration via GLOBAL_LOAD_ASYNC_TO_LDS_B128, overlapping the next tile's
// fetch with the current tile's compute. Each wave reads only LDS bytes it
// staged itself -> ASYNCcnt is the only sync needed (no barriers).
// Requires: nPairs divisible by gridDim*blockDim (launcher guarantees).
// ---------------------------------------------------------------------------
__global__ __launch_bounds__(256) void net24_async_kernel(
    const float4* __restrict__ x4,        // 2 rows per float4
    float2* __restrict__ out2,            // 2 outputs per float2
    const float* __restrict__ fc1_tw, const float* __restrict__ fc1_pw,
    const float* __restrict__ fc1_b,  const float* __restrict__ m4_tw,
    const float* __restrict__ m4_pw,  const float* __restrict__ m4_b,
    int iters, int strideP) {
    __shared__ float4 buf[2][256];

    const NetP s = load_params(fc1_tw, fc1_pw, fc1_b, m4_tw, m4_pw, m4_b);
    const int tid = threadIdx.x;
    int i = blockIdx.x * 256 + tid;

    const uint64_t gbase = (uint64_t)(uintptr_t)x4;
    // Generic pointers to LDS carry the LDS byte offset in their low 32 bits.
    const uint32_t lds0 = (uint32_t)(uintptr_t)&buf[0][tid];
    const uint32_t lds1 = (uint32_t)(uintptr_t)&buf[1][tid];

    // Prime the pipeline: stage tile 0 into buffer 0.
    asm volatile("global_load_async_to_lds_b128 %0, %1, %2"
                 :: "v"(lds0), "v"((uint32_t)i * 16u), "s"(gbase)
                 : "memory");

    int par = 0;
    for (int k = 0; k < iters; ++k) {
        const int inext = i + strideP;
        if (k + 1 < iters) {
            // Kick off the next tile into the other buffer, then wait until
            // only that one is outstanding (oldest completes in order).
            const uint32_t ln = par ? lds0 : lds1;
            asm volatile("global_load_async_to_lds_b128 %0, %1, %2"
                         :: "v"(ln), "v"((uint32_t)inext * 16u), "s"(gbase)
                         : "memory");
            asm volatile("s_wait_asynccnt 0x1" ::: "memory");
        } else {
            asm volatile("s_wait_asynccnt 0x0" ::: "memory");
        }

        const float4 xv = buf[par][tid];   // ds_load_b128
        float2 o;
        o.x = net_row(xv.x, xv.y, s);
        o.y = net_row(xv.z, xv.w, s);
        out2[i] = o;                       // global_store_b64

        i = inext;
        par ^= 1;
    }
}

// Fallback: plain grid-stride, one row per lane, any size.
__global__ __launch_bounds__(256) void net24_plain_kernel(
    const float2* __restrict__ x2, float* __restrict__ out,
    const float* __restrict__ fc1_tw, const float* __restrict__ fc1_pw,
    const float* __restrict__ fc1_b,  const float* __restrict__ m4_tw,
    const float* __restrict__ m4_pw,  const float* __restrict__ m4_b,
    int nRows) {
    const NetP s = load_params(fc1_tw, fc1_pw, fc1_b, m4_tw, m4_pw, m4_b);
    int i = blockIdx.x * blockDim.x + threadIdx.x;
    const int stride = gridDim.x * blockDim.x;
    for (; i < nRows; i += stride) {
        const float2 xv = x2[i];
        out[i] = net_row(xv.x, xv.y, s);
    }
}

extern "C" void kernel_launch(void* const* d_in, const int* in_sizes, int n_in,
                              void* d_out, int out_size, void* d_ws, size_t ws_size,
                              hipStream_t stream) {
    const float* x      = (const float*)d_in[0];
    const float* fc1_tw = (const float*)d_in[1];
    const float* fc1_pw = (const float*)d_in[2];
    const float* fc1_b  = (const float*)d_in[3];
    const float* m4_tw  = (const float*)d_in[4];
    const float* m4_pw  = (const float*)d_in[5];
    const float* m4_b   = (const float*)d_in[6];

    const int nRows = out_size;                 // B (output shape (B,1))
    const int block = 256;                      // 8 wave32 per block
    const int grid  = 2048;
    const long totalThreads = (long)block * grid;
    const long nPairs = (long)nRows / 2;

    if ((nRows & 1) == 0 && nPairs % totalThreads == 0) {
        const int iters = (int)(nPairs / totalThreads);   // 8 for B = 8388608
        net24_async_kernel<<<grid, block, 0, stream>>>(
            (const float4*)x, (float2*)d_out,
            fc1_tw, fc1_pw, fc1_b, m4_tw, m4_pw, m4_b,
            iters, (int)totalThreads);
    } else {
        int g = (int)((nRows + block - 1) / block);
        if (g > 16384) g = 16384;
        if (g < 1) g = 1;
        net24_plain_kernel<<<g, block, 0, stream>>>(
            (const float2*)x, (float*)d_out,
            fc1_tw, fc1_pw, fc1_b, m4_tw, m4_pw, m4_b, nRows);
    }
}